// SparseLinear_41755672052511
// MI455X (gfx1250) — compile-verified
//
#include <hip/hip_runtime.h>
#include <hip/hip_bf16.h>
#include <stdint.h>

// COO SpMM, out[m,n] = sum_{nnz} val * x[col,n], binned by 16-row tile and
// executed as rank-4 WMMA updates (V_WMMA_F32_16X16X4_F32) with register-
// resident 16x16 f32 accumulators. Pipeline: zero -> count -> scan -> scatter
// -> wmma-accumulate. Each wave owns 4 N-tiles so the per-chunk metadata
// decode + A-matrix build is amortized over 4 WMMAs, which also gives 4
// independent accumulator chains (WMMA ILP) without extra unrolling.
// Working set (~56MB) is L2-resident (192MB): gather-throughput bound.

#define M_DIM   16384
#define K_DIM   16384
#define N_DIM   128
#define NNZ_CNT 2097152
#define TILE_M  16
#define NBINS   (M_DIM / TILE_M)   // 1024
#define SEG     8                  // nnz-segments per bin (occupancy)
#define TPW     4                  // N-tiles per wave (64 columns)

typedef __attribute__((ext_vector_type(2))) float v2f;
typedef __attribute__((ext_vector_type(8))) float v8f;

__global__ void k_zero(float* __restrict__ out, unsigned* __restrict__ counts) {
  int i = blockIdx.x * blockDim.x + threadIdx.x;
  if (i < M_DIM * N_DIM) out[i] = 0.0f;
  if (i < NBINS) counts[i] = 0u;
}

__global__ void k_count(const int* __restrict__ rows, unsigned* __restrict__ counts) {
  int i = blockIdx.x * blockDim.x + threadIdx.x;
  if (i < NNZ_CNT) atomicAdd(&counts[((unsigned)rows[i]) >> 4], 1u);
}

// single block, 1024 threads: Hillis-Steele inclusive scan -> exclusive offsets
__global__ void __launch_bounds__(NBINS)
k_scan(const unsigned* __restrict__ counts, unsigned* __restrict__ offsets,
       unsigned* __restrict__ cursor) {
  __shared__ unsigned s[NBINS];
  int t = threadIdx.x;
  s[t] = counts[t];
  for (int off = 1; off < NBINS; off <<= 1) {
    __syncthreads();
    unsigned v = (t >= off) ? s[t - off] : 0u;
    __syncthreads();
    s[t] += v;
  }
  __syncthreads();
  unsigned excl = (t == 0) ? 0u : s[t - 1];
  offsets[t] = excl;
  cursor[t]  = excl;
  if (t == NBINS - 1) offsets[NBINS] = s[NBINS - 1];
}

// scatter nonzeros into their bin; pack col (18 bits) + local row (4 bits)
__global__ void k_scatter(const float* __restrict__ values, const int* __restrict__ rows,
                          const int* __restrict__ cols, unsigned* __restrict__ cursor,
                          float* __restrict__ sv, unsigned* __restrict__ scv) {
  int i = blockIdx.x * blockDim.x + threadIdx.x;
  if (i >= NNZ_CNT) return;
  unsigned r = (unsigned)rows[i];
  unsigned b = r >> 4;
  unsigned p = atomicAdd(&cursor[b], 1u);
  sv[p]  = values[i];
  scv[p] = ((unsigned)cols[i]) | ((r & 15u) << 18);
}

// ---- one chunk of 4 nonzeros: build A once, feed TPW WMMAs ----
// A 16x4 f32 layout: lanes0-15 VGPR0=K0,VGPR1=K1; lanes16-31 VGPR0=K2,VGPR1=K3
// B 4x16  f32 layout: VGPR0 rows K0(lanes0-15)/K1(lanes16-31), VGPR1 K2/K3
__device__ __forceinline__ void wmma_chunk(v8f acc[TPW], const float* __restrict__ x,
                                           float v0, float v1, float v2, float v3,
                                           unsigned c0, unsigned c1, unsigned c2, unsigned c3,
                                           int half, int lrow, int ncol0) {
  const unsigned col0 = c0 & 0x3FFFFu, r0 = c0 >> 18;
  const unsigned col1 = c1 & 0x3FFFFu, r1 = c1 >> 18;
  const unsigned col2 = c2 & 0x3FFFFu, r2 = c2 >> 18;
  const unsigned col3 = c3 & 0x3FFFFu, r3 = c3 >> 18;

  const float    va = half ? v2 : v0;  const unsigned ra = half ? r2 : r0;
  const float    vb = half ? v3 : v1;  const unsigned rb = half ? r3 : r1;
  v2f A;
  A.x = (ra == (unsigned)lrow) ? va : 0.f;
  A.y = (rb == (unsigned)lrow) ? vb : 0.f;

  const unsigned bc0 = half ? col1 : col0;
  const unsigned bc1 = half ? col3 : col2;
  const float* rp0 = x + (size_t)bc0 * N_DIM + ncol0;  // K0/K1 row slice base
  const float* rp1 = x + (size_t)bc1 * N_DIM + ncol0;  // K2/K3 row slice base

#pragma unroll
  for (int j = 0; j < TPW; ++j) {
    v2f B;
    B.x = rp0[16 * j];   // shared base + immediate offset -> clause-able loads
    B.y = rp1[16 * j];
    acc[j] = __builtin_amdgcn_wmma_f32_16x16x4_f32(
        /*neg_a=*/false, A, /*neg_b=*/false, B,
        /*c_mod=*/(short)0, acc[j], /*reuse_a=*/false, /*reuse_b=*/false);
  }
}

// Block = 64 threads = 2 waves; wave w covers columns [w*64, w*64+64).
// Grid = NBINS*SEG blocks; each block processes one (bin, nnz-segment).
__global__ void __launch_bounds__(64)
k_wmma(const float* __restrict__ x, const float* __restrict__ sv,
       const unsigned* __restrict__ scv, const unsigned* __restrict__ offsets,
       float* __restrict__ out) {
  const int bin   = blockIdx.x / SEG;
  const int seg   = blockIdx.x % SEG;
  const int wave  = threadIdx.x >> 5;   // 0..1
  const int lane  = threadIdx.x & 31;
  const int half  = lane >> 4;          // 0: lanes 0-15, 1: lanes 16-31
  const int lrow  = lane & 15;
  const int ncol0 = wave * (TPW * 16) + lrow;  // first of this lane's 4 columns

  const unsigned s = offsets[bin];
  const unsigned e = offsets[bin + 1];
  const unsigned len    = e - s;
  const unsigned segLen = (len + SEG - 1) / SEG;
  const unsigned p0   = s + (unsigned)seg * segLen;
  const unsigned pend = (p0 + segLen < e) ? (p0 + segLen) : e;
  const unsigned n    = (pend > p0) ? (pend - p0) : 0u;

  v8f acc[TPW];
#pragma unroll
  for (int j = 0; j < TPW; ++j) acc[j] = (v8f){0.f,0.f,0.f,0.f,0.f,0.f,0.f,0.f};

  // ---- main loop: one full chunk of 4 nnz per iteration, no guards ----
  unsigned p = p0;
  const unsigned pfull = p0 + (n & ~3u);
  for (; p < pfull; p += 4) {
    float    v0 = sv[p],  v1 = sv[p + 1],  v2 = sv[p + 2],  v3 = sv[p + 3];
    unsigned c0 = scv[p], c1 = scv[p + 1], c2 = scv[p + 2], c3 = scv[p + 3];
    wmma_chunk(acc, x, v0, v1, v2, v3, c0, c1, c2, c3, half, lrow, ncol0);
  }
  // ---- tail (0-3 nnz), guards run exactly once ----
  if (p < pend) {
    float    v0 = sv[p];  unsigned c0 = scv[p];
    float    v1 = 0.f, v2 = 0.f, v3 = 0.f;
    unsigned c1 = 0u,  c2 = 0u,  c3 = 0u;
    if (p + 1 < pend) { v1 = sv[p + 1]; c1 = scv[p + 1]; }
    if (p + 2 < pend) { v2 = sv[p + 2]; c2 = scv[p + 2]; }
    wmma_chunk(acc, x, v0, v1, v2, v3, c0, c1, c2, c3, half, lrow, ncol0);
  }

  // C/D layout: VGPR v -> M = v (lanes0-15) / v+8 (lanes16-31), N = lane%16
  float* obase = out + (size_t)(bin * TILE_M + 8 * half) * N_DIM + ncol0;
#pragma unroll
  for (int j = 0; j < TPW; ++j) {
#pragma unroll
    for (int v = 0; v < 8; ++v) {
      unsafeAtomicAdd(obase + (size_t)v * N_DIM + 16 * j, acc[j][v]);
    }
  }
}

extern "C" void kernel_launch(void* const* d_in, const int* in_sizes, int n_in,
                              void* d_out, int out_size, void* d_ws, size_t ws_size,
                              hipStream_t stream) {
  const float* x      = (const float*)d_in[0];
  const float* values = (const float*)d_in[1];
  const int*   rows   = (const int*)d_in[2];
  const int*   cols   = (const int*)d_in[3];
  float*       out    = (float*)d_out;

  char* ws = (char*)d_ws;
  unsigned* counts  = (unsigned*)(ws + 0);        // 1024 u32
  unsigned* offsets = (unsigned*)(ws + 4096);     // 1025 u32
  unsigned* cursor  = (unsigned*)(ws + 8192);     // 1024 u32
  float*    sv      = (float*)   (ws + 16384);                     // NNZ f32
  unsigned* scv     = (unsigned*)(ws + 16384 + (size_t)NNZ_CNT*4); // NNZ u32

  const int zb = (M_DIM * N_DIM + 255) / 256;      // 8192
  const int nb = (NNZ_CNT + 255) / 256;            // 8192

  k_zero   <<<zb, 256, 0, stream>>>(out, counts);
  k_count  <<<nb, 256, 0, stream>>>(rows, counts);
  k_scan   <<<1, NBINS, 0, stream>>>(counts, offsets, cursor);
  k_scatter<<<nb, 256, 0, stream>>>(values, rows, cols, cursor, sv, scv);
  k_wmma   <<<NBINS * SEG, 64, 0, stream>>>(x, sv, scv, offsets, out);
}